// QWenAttention_6287832121706
// MI455X (gfx1250) — compile-verified
//
#include <hip/hip_runtime.h>
#include <cstdint>

typedef __attribute__((ext_vector_type(2))) float v2f;
typedef __attribute__((ext_vector_type(8))) float v8f;
typedef __attribute__((ext_vector_type(4))) unsigned int v4u;
typedef __attribute__((ext_vector_type(4))) int v4i;
typedef __attribute__((ext_vector_type(8))) int v8i;

#define S_LEN 2048
#define HID   4096
#define NH    32
#define HD    128
#define QKV_N (3 * HID)
#define SCALE 0.08838834764831845f  // 1/sqrt(128)

#if defined(__HIP_DEVICE_COMPILE__) && __has_builtin(__builtin_amdgcn_tensor_load_to_lds)
#define KTDM 1
#else
#define KTDM 0
#endif

// ---------------------------------------------------------------------------
// TDM helper: DMA a 2D fp32 tile (tile_w x tile_h) from global memory into LDS
// with row padding (CDNA5 ISA). Issued per-wave; tracked on
// TENSORcnt. pad_int_code: interval = 2^(code+1) DWORDs; pad_amt_code: code+1
// DWORDs appended per interval (36-float rows: 4/3, 132-float rows: 6/3).
// ---------------------------------------------------------------------------
__device__ __forceinline__ unsigned lds_off(const void* p) {
  // low 32 bits of a generic LDS address == byte offset into LDS aperture
  return (unsigned)(uintptr_t)p;
}

__device__ __forceinline__ void tdm_load_tile_f32(unsigned lds_byte,
                                                  const float* gptr,
                                                  unsigned tile_w, unsigned tile_h,
                                                  unsigned row_stride,
                                                  unsigned pad_int_code,
                                                  unsigned pad_amt_code) {
#if KTDM
  unsigned long long ga = (unsigned long long)(uintptr_t)gptr;
  // D# group0: count=1, lds_addr, 57-bit global addr, type=2 ("image")
  v4u g0 = { 0x1u,
             lds_byte,
             (unsigned)(ga & 0xffffffffu),
             (unsigned)((ga >> 32) & 0x1ffffffu) | (2u << 30) };
  // D# group1: data_size=4B(2), pad_enable, pad codes; tensor dims == tile dims
  unsigned w0 = (2u << 16) | (1u << 20) | (pad_int_code << 22) | (pad_amt_code << 25);
  v8i g1 = { (int)w0,
             (int)((tile_w & 0xffffu) << 16),   // [31:16] tensor_dim0 lo
             (int)((tile_h & 0xffffu) << 16),   // [15:0] dim0 hi=0, [31:16] tensor_dim1 lo
             (int)((tile_w & 0xffffu) << 16),   // [15:0] dim1 hi=0, [31:16] tile_dim0
             (int)(tile_h & 0xffffu),           // [15:0] tile_dim1, [31:16] tile_dim2=0
             (int)row_stride,                   // tensor_dim0_stride[31:0]
             0, 0 };
  v4i gz = { 0, 0, 0, 0 };
#if __clang_major__ >= 23
  v8i gz8 = { 0, 0, 0, 0, 0, 0, 0, 0 };
  __builtin_amdgcn_tensor_load_to_lds(g0, g1, gz, gz, gz8, 0);
#else
  __builtin_amdgcn_tensor_load_to_lds(g0, g1, gz, gz, 0);
#endif
#else
  (void)lds_byte; (void)gptr; (void)tile_w; (void)tile_h;
  (void)row_stride; (void)pad_int_code; (void)pad_amt_code;
#endif
}

__device__ __forceinline__ void tdm_wait_all() {
#if KTDM
  __builtin_amdgcn_s_wait_tensorcnt(0);
#endif
}

// ---------------------------------------------------------------------------
// Shared WMMA inner chunk: 32-deep K slice, 2 output tiles per wave.
// f32 16x16x4 fragment layout: A lane=M (0..15 twice), VGPR pair holds K=koff,
// koff+1 with koff = 2*(lane>>4); B mirrored with lane=N.
// ---------------------------------------------------------------------------
__device__ __forceinline__ void wmma_chunk(const float (*As)[36],
                                           const float (*Bs)[132],
                                           int tm, int tn0, int l16, int koff,
                                           v8f& acc0, v8f& acc1) {
#pragma unroll
  for (int kk = 0; kk < 32; kk += 4) {
    v2f a, b0, b1;
    a.x  = As[tm * 16 + l16][kk + koff];
    a.y  = As[tm * 16 + l16][kk + koff + 1];
    b0.x = Bs[kk + koff][tn0 * 16 + l16];
    b0.y = Bs[kk + koff + 1][tn0 * 16 + l16];
    b1.x = Bs[kk + koff][(tn0 + 1) * 16 + l16];
    b1.y = Bs[kk + koff + 1][(tn0 + 1) * 16 + l16];
    acc0 = __builtin_amdgcn_wmma_f32_16x16x4_f32(false, a, false, b0, (short)0, acc0, false, false);
    acc1 = __builtin_amdgcn_wmma_f32_16x16x4_f32(false, a, false, b1, (short)0, acc1, false, false);
  }
}

// ---------------------------------------------------------------------------
// Kernel 1: QKV GEMM (+bias) with fused RoPE epilogue.
// TDM double-buffered tile streaming: wave0 issues next A/B tile DMA while all
// 8 waves run WMMA on the current buffer.
// ---------------------------------------------------------------------------
__launch_bounds__(256)
__global__ void qkv_rope_kernel(const float* __restrict__ x,
                                const float* __restrict__ w,
                                const float* __restrict__ bias,
                                float* __restrict__ q_out,
                                float* __restrict__ k_out,
                                float* __restrict__ v_out) {
#if KTDM
  __shared__ float As[2][32][36];
  __shared__ float Bs[2][32][132];
#else
  __shared__ float As[1][32][36];
  __shared__ float Bs[1][32][132];
#endif
  __shared__ float Cs[32][132];

  const int n0   = blockIdx.x * 128;   // column base in [0, 12288)
  const int m0   = blockIdx.y * 32;    // row base in [0, 2048)
  const int t    = threadIdx.x;
  const int wv   = t >> 5;
  const int lane = t & 31;
  const int l16  = lane & 15;
  const int koff = (lane >> 4) << 1;

  const int tm  = wv >> 2;             // 0..1
  const int tn0 = (wv & 3) * 2;        // 0,2,4,6

  v8f acc0 = {};
  v8f acc1 = {};

#if KTDM
  if (wv == 0) {
    tdm_load_tile_f32(lds_off(&As[0][0][0]), x + (size_t)m0 * HID, 32, 32, HID, 4, 3);
    tdm_load_tile_f32(lds_off(&Bs[0][0][0]), w + n0, 128, 32, QKV_N, 6, 3);
  }
  for (int k0 = 0; k0 < HID; k0 += 32) {
    const int b = (k0 >> 5) & 1;
    if (wv == 0) tdm_wait_all();
    __syncthreads();
    if (wv == 0 && (k0 + 32) < HID) {
      tdm_load_tile_f32(lds_off(&As[b ^ 1][0][0]),
                        x + (size_t)m0 * HID + k0 + 32, 32, 32, HID, 4, 3);
      tdm_load_tile_f32(lds_off(&Bs[b ^ 1][0][0]),
                        w + (size_t)(k0 + 32) * QKV_N + n0, 128, 32, QKV_N, 6, 3);
    }
    wmma_chunk(As[b], Bs[b], tm, tn0, l16, koff, acc0, acc1);
    __syncthreads();
  }
#else
  for (int k0 = 0; k0 < HID; k0 += 32) {
    {
      int row = t >> 3, c4 = (t & 7) * 4;
      float4 av = *(const float4*)(x + (size_t)(m0 + row) * HID + k0 + c4);
      As[0][row][c4 + 0] = av.x; As[0][row][c4 + 1] = av.y;
      As[0][row][c4 + 2] = av.z; As[0][row][c4 + 3] = av.w;
    }
    {
      int c4 = (t & 31) * 4;
#pragma unroll
      for (int i = 0; i < 4; ++i) {
        int row = i * 8 + (t >> 5);
        float4 bv = *(const float4*)(w + (size_t)(k0 + row) * QKV_N + n0 + c4);
        Bs[0][row][c4 + 0] = bv.x; Bs[0][row][c4 + 1] = bv.y;
        Bs[0][row][c4 + 2] = bv.z; Bs[0][row][c4 + 3] = bv.w;
      }
    }
    __syncthreads();
    wmma_chunk(As[0], Bs[0], tm, tn0, l16, koff, acc0, acc1);
    __syncthreads();
  }
#endif

  // spill C to LDS with bias added
#pragma unroll
  for (int i = 0; i < 8; ++i) {
    int m  = tm * 16 + i + ((lane >> 4) << 3);
    int na = tn0 * 16 + l16;
    int nb = (tn0 + 1) * 16 + l16;
    Cs[m][na] = acc0[i] + bias[n0 + na];
    Cs[m][nb] = acc1[i] + bias[n0 + nb];
  }
  __syncthreads();

  // RoPE epilogue (q,k) or passthrough (v)
  const int sec  = n0 / HID;           // 0=q 1=k 2=v
  const int head = (n0 % HID) / HD;
  float* outp = (sec == 0) ? q_out : (sec == 1) ? k_out : v_out;

#pragma unroll
  for (int it = 0; it < 16; ++it) {
    int idx = t + it * 256;
    int m = idx >> 7;
    int d = idx & 127;
    int s = m0 + m;
    float val = Cs[m][d];
    float res;
    if (sec < 2) {
      int p = d & 63;
      float inv   = __powf(10000.0f, -(float)p * (1.0f / 64.0f));
      float theta = (float)s * inv;
      float sn, cs;
      __sincosf(theta, &sn, &cs);
      float rot = (d < 64) ? -Cs[m][d + 64] : Cs[m][d - 64];
      res = val * cs + rot * sn;
    } else {
      res = val;
    }
    outp[((size_t)head * S_LEN + s) * HD + d] = res;
  }
}

// ---------------------------------------------------------------------------
// Kernel 2: causal flash attention, fp32 WMMA, online softmax.
// 128 threads (4 waves); Q block resident; K/V tiles DMA'd per block by TDM.
// ---------------------------------------------------------------------------
__launch_bounds__(128)
__global__ void flash_attn_kernel(const float* __restrict__ q,
                                  const float* __restrict__ k,
                                  const float* __restrict__ v,
                                  float* __restrict__ o /* [S][HID] */) {
  __shared__ float Qs[32][132];
  __shared__ float Ks[32][132];
  __shared__ float Vs[32][132];
  __shared__ float Ss[32][36];
  __shared__ float row_m[32], row_l[32], row_a[32];

  const int head = blockIdx.y;
  const int qb   = blockIdx.x;
  const int q0   = qb * 32;
  const int t    = threadIdx.x;
  const int wv   = t >> 5;
  const int lane = t & 31;
  const int l16  = lane & 15;
  const int koff = (lane >> 4) << 1;

  const float* qh = q + (size_t)head * S_LEN * HD;
  const float* kh = k + (size_t)head * S_LEN * HD;
  const float* vh = v + (size_t)head * S_LEN * HD;

#if KTDM
  if (wv == 0) {  // Q block DMA; completion covered by first in-loop wait
    tdm_load_tile_f32(lds_off(&Qs[0][0]), qh + (size_t)q0 * HD, 128, 32, HD, 6, 3);
  }
#else
#pragma unroll
  for (int i = 0; i < 8; ++i) {
    int f4 = t + i * 128;
    int row = f4 >> 5;
    int c4 = (f4 & 31) * 4;
    float4 qv = *(const float4*)(qh + (size_t)(q0 + row) * HD + c4);
    Qs[row][c4 + 0] = qv.x; Qs[row][c4 + 1] = qv.y;
    Qs[row][c4 + 2] = qv.z; Qs[row][c4 + 3] = qv.w;
  }
#endif
  if (t < 32) { row_m[t] = -3.0e38f; row_l[t] = 0.0f; }

  v8f oacc[2][2] = {};
  __syncthreads();

  for (int kb = 0; kb <= qb; ++kb) {
    const int kk0 = kb * 32;
#if KTDM
    if (wv == 0) {
      tdm_load_tile_f32(lds_off(&Ks[0][0]), kh + (size_t)kk0 * HD, 128, 32, HD, 6, 3);
      tdm_load_tile_f32(lds_off(&Vs[0][0]), vh + (size_t)kk0 * HD, 128, 32, HD, 6, 3);
      tdm_wait_all();
    }
#else
#pragma unroll
    for (int i = 0; i < 8; ++i) {
      int f4 = t + i * 128;
      int row = f4 >> 5;
      int c4 = (f4 & 31) * 4;
      float4 kv = *(const float4*)(kh + (size_t)(kk0 + row) * HD + c4);
      Ks[row][c4 + 0] = kv.x; Ks[row][c4 + 1] = kv.y;
      Ks[row][c4 + 2] = kv.z; Ks[row][c4 + 3] = kv.w;
      float4 vv = *(const float4*)(vh + (size_t)(kk0 + row) * HD + c4);
      Vs[row][c4 + 0] = vv.x; Vs[row][c4 + 1] = vv.y;
      Vs[row][c4 + 2] = vv.z; Vs[row][c4 + 3] = vv.w;
    }
#endif
    __syncthreads();

    // scores: wave wv -> 16x16 tile (wv>>1, wv&1); K-dim 128
    {
      int tm = wv >> 1, tn = wv & 1;
      v8f sc = {};
#pragma unroll
      for (int kk = 0; kk < HD; kk += 4) {
        v2f a, b;
        a.x = Qs[tm * 16 + l16][kk + koff];
        a.y = Qs[tm * 16 + l16][kk + koff + 1];
        b.x = Ks[tn * 16 + l16][kk + koff];       // B[k][n] = K[n][k]
        b.y = Ks[tn * 16 + l16][kk + koff + 1];
        sc = __builtin_amdgcn_wmma_f32_16x16x4_f32(false, a, false, b, (short)0, sc, false, false);
      }
#pragma unroll
      for (int i = 0; i < 8; ++i) {
        int m = tm * 16 + i + ((lane >> 4) << 3);
        int n = tn * 16 + l16;
        int gq = q0 + m, gk = kk0 + n;
        Ss[m][n] = (gk <= gq) ? sc[i] * SCALE : -3.0e38f;
      }
    }
    __syncthreads();

    // online softmax stats, one thread per query row
    if (t < 32) {
      float mx = row_m[t];
#pragma unroll
      for (int c = 0; c < 32; ++c) mx = fmaxf(mx, Ss[t][c]);
      float alpha = __expf(row_m[t] - mx);
      float sum = 0.0f;
#pragma unroll
      for (int c = 0; c < 32; ++c) {
        float p = __expf(Ss[t][c] - mx);
        Ss[t][c] = p;
        sum += p;
      }
      row_m[t] = mx;
      row_l[t] = row_l[t] * alpha + sum;
      row_a[t] = alpha;
    }
    __syncthreads();

    // rescale O and accumulate P@V (K-dim = 32 key positions)
#pragma unroll
    for (int tm = 0; tm < 2; ++tm) {
#pragma unroll
      for (int tn = 0; tn < 2; ++tn) {
        v8f acc = oacc[tm][tn];
        int m_base = tm * 16 + ((lane >> 4) << 3);
#pragma unroll
        for (int i = 0; i < 8; ++i) acc[i] *= row_a[m_base + i];
        int vn = wv * 32 + tn * 16 + l16;
#pragma unroll
        for (int kk = 0; kk < 32; kk += 4) {
          v2f a, b;
          a.x = Ss[tm * 16 + l16][kk + koff];
          a.y = Ss[tm * 16 + l16][kk + koff + 1];
          b.x = Vs[kk + koff][vn];
          b.y = Vs[kk + koff + 1][vn];
          acc = __builtin_amdgcn_wmma_f32_16x16x4_f32(false, a, false, b, (short)0, acc, false, false);
        }
        oacc[tm][tn] = acc;
      }
    }
    __syncthreads();
  }

  // finalize: divide by running sum, write [S][HID]
#pragma unroll
  for (int tm = 0; tm < 2; ++tm) {
#pragma unroll
    for (int tn = 0; tn < 2; ++tn) {
#pragma unroll
      for (int i = 0; i < 8; ++i) {
        int m = tm * 16 + i + ((lane >> 4) << 3);
        int dcol = wv * 32 + tn * 16 + l16;
        int s = q0 + m;
        o[(size_t)s * HID + head * HD + dcol] = oacc[tm][tn][i] / row_l[m];
      }
    }
  }
}

// ---------------------------------------------------------------------------
// Kernel 3: output projection GEMM 2048x4096x4096 -> d_out (TDM streamed)
// ---------------------------------------------------------------------------
__launch_bounds__(256)
__global__ void proj_kernel(const float* __restrict__ a_in,
                            const float* __restrict__ w,
                            float* __restrict__ out) {
#if KTDM
  __shared__ float As[2][32][36];
  __shared__ float Bs[2][32][132];
#else
  __shared__ float As[1][32][36];
  __shared__ float Bs[1][32][132];
#endif

  const int n0   = blockIdx.x * 128;
  const int m0   = blockIdx.y * 32;
  const int t    = threadIdx.x;
  const int wv   = t >> 5;
  const int lane = t & 31;
  const int l16  = lane & 15;
  const int koff = (lane >> 4) << 1;

  const int tm  = wv >> 2;
  const int tn0 = (wv & 3) * 2;

  v8f acc0 = {};
  v8f acc1 = {};

#if KTDM
  if (wv == 0) {
    tdm_load_tile_f32(lds_off(&As[0][0][0]), a_in + (size_t)m0 * HID, 32, 32, HID, 4, 3);
    tdm_load_tile_f32(lds_off(&Bs[0][0][0]), w + n0, 128, 32, HID, 6, 3);
  }
  for (int k0 = 0; k0 < HID; k0 += 32) {
    const int b = (k0 >> 5) & 1;
    if (wv == 0) tdm_wait_all();
    __syncthreads();
    if (wv == 0 && (k0 + 32) < HID) {
      tdm_load_tile_f32(lds_off(&As[b ^ 1][0][0]),
                        a_in + (size_t)m0 * HID + k0 + 32, 32, 32, HID, 4, 3);
      tdm_load_tile_f32(lds_off(&Bs[b ^ 1][0][0]),
                        w + (size_t)(k0 + 32) * HID + n0, 128, 32, HID, 6, 3);
    }
    wmma_chunk(As[b], Bs[b], tm, tn0, l16, koff, acc0, acc1);
    __syncthreads();
  }
#else
  for (int k0 = 0; k0 < HID; k0 += 32) {
    {
      int row = t >> 3, c4 = (t & 7) * 4;
      float4 av = *(const float4*)(a_in + (size_t)(m0 + row) * HID + k0 + c4);
      As[0][row][c4 + 0] = av.x; As[0][row][c4 + 1] = av.y;
      As[0][row][c4 + 2] = av.z; As[0][row][c4 + 3] = av.w;
    }
    {
      int c4 = (t & 31) * 4;
#pragma unroll
      for (int i = 0; i < 4; ++i) {
        int row = i * 8 + (t >> 5);
        float4 bv = *(const float4*)(w + (size_t)(k0 + row) * HID + n0 + c4);
        Bs[0][row][c4 + 0] = bv.x; Bs[0][row][c4 + 1] = bv.y;
        Bs[0][row][c4 + 2] = bv.z; Bs[0][row][c4 + 3] = bv.w;
      }
    }
    __syncthreads();
    wmma_chunk(As[0], Bs[0], tm, tn0, l16, koff, acc0, acc1);
    __syncthreads();
  }
#endif

#pragma unroll
  for (int i = 0; i < 8; ++i) {
    int m = tm * 16 + i + ((lane >> 4) << 3);
    out[(size_t)(m0 + m) * HID + n0 + tn0 * 16 + l16]       = acc0[i];
    out[(size_t)(m0 + m) * HID + n0 + (tn0 + 1) * 16 + l16] = acc1[i];
  }
}

// ---------------------------------------------------------------------------
extern "C" void kernel_launch(void* const* d_in, const int* in_sizes, int n_in,
                              void* d_out, int out_size, void* d_ws, size_t ws_size,
                              hipStream_t stream) {
  (void)in_sizes; (void)n_in; (void)out_size; (void)ws_size;
  const float* x      = (const float*)d_in[0];
  const float* w_attn = (const float*)d_in[1];
  const float* b_attn = (const float*)d_in[2];
  const float* w_proj = (const float*)d_in[3];
  float* out = (float*)d_out;

  const size_t per = (size_t)NH * S_LEN * HD;  // 8M floats
  float* q_ws = (float*)d_ws;
  float* k_ws = q_ws + per;
  float* v_ws = k_ws + per;
  float* a_ws = v_ws + per;                    // [S][HID]

  qkv_rope_kernel<<<dim3(QKV_N / 128, S_LEN / 32), 256, 0, stream>>>(
      x, w_attn, b_attn, q_ws, k_ws, v_ws);
  flash_attn_kernel<<<dim3(S_LEN / 32, NH), 128, 0, stream>>>(
      q_ws, k_ws, v_ws, a_ws);
  proj_kernel<<<dim3(HID / 128, S_LEN / 32), 256, 0, stream>>>(
      a_ws, w_proj, out);
}